// VQVAEModel_79680233275699
// MI455X (gfx1250) — compile-verified
//
#include <hip/hip_runtime.h>
#include <hip/hip_bf16.h>

typedef __attribute__((ext_vector_type(2))) float v2f;
typedef __attribute__((ext_vector_type(8))) float v8f;
typedef int v4i __attribute__((vector_size(16)));   // matches builtin's param type

#define DIM        256
#define NCODES     4096
#define NROWS      32768
#define ROWS_PER_BLOCK 64          // 4 waves x 16 rows
#define BTILE      16              // codes per LDS stage

// ---------------------------------------------------------------------------
// Async global->LDS plumbing (CDNA5 GLOBAL_LOAD_ASYNC_TO_LDS_B128, ASYNCcnt).
// Builtin signature (probe-confirmed): (AS1 v4i* src, AS3 v4i* dst, imm, imm)
// ---------------------------------------------------------------------------
#if __has_builtin(__builtin_amdgcn_global_load_async_to_lds_b128)
#define VQ_ASYNC 1
#else
#define VQ_ASYNC 0
#endif

__device__ __forceinline__ void async_copy_b128(const float* gsrc, float* ldst) {
#if VQ_ASYNC
    typedef __attribute__((address_space(1))) v4i gv4i;
    typedef __attribute__((address_space(3))) v4i lv4i;
    gv4i* g = (gv4i*)(unsigned long long)gsrc;
    lv4i* l = (lv4i*)(unsigned int)(unsigned long long)ldst;  // low 32 bits = LDS offset
    __builtin_amdgcn_global_load_async_to_lds_b128(g, l, 0, 0);
#else
    *(float4*)ldst = *(const float4*)gsrc;
#endif
}

template <int N>
__device__ __forceinline__ void wait_asynccnt() {
#if VQ_ASYNC
#if __has_builtin(__builtin_amdgcn_s_wait_asynccnt)
    __builtin_amdgcn_s_wait_asynccnt(N);
#else
    asm volatile("s_wait_asynccnt %0" ::"i"(N) : "memory");
#endif
#endif
}

// ---------------------------------------------------------------------------
// Kernel 1: ||e_k||^2 for every codebook entry
// ---------------------------------------------------------------------------
__global__ void __launch_bounds__(256)
vq_enorm_kernel(const float* __restrict__ E, float* __restrict__ enorm) {
    int k = blockIdx.x * blockDim.x + threadIdx.x;
    if (k >= NCODES) return;
    const float4* row = (const float4*)(E + (size_t)k * DIM);
    float s = 0.f;
#pragma unroll 4
    for (int j = 0; j < DIM / 4; ++j) {
        float4 v = row[j];
        s += v.x * v.x + v.y * v.y + v.z * v.z + v.w * v.w;
    }
    enorm[k] = s;
}

// ---------------------------------------------------------------------------
// Kernel 2: fused GEMM (x . e^T via V_WMMA_F32_16X16X4_F32) + running argmin.
// Block: 128 threads = 4 waves; wave w owns rows [blk*64 + w*16, +16).
// LDS: A tile 64x256 f32 (async-staged once) +
//      double-buffered B tile 2 x (16x256) f32 (async pipeline, ASYNCcnt).
// K-tile loop unrolled by 2 so each stage uses a STATIC buffer base pointer
// -> inner loops are pure ds_load_2addr_b64 + v_wmma, no address ALU.
// ---------------------------------------------------------------------------
__global__ void __launch_bounds__(128)
vq_argmin_kernel(const float* __restrict__ X, const float* __restrict__ E,
                 const float* __restrict__ enorm, int* __restrict__ idx_out) {
    extern __shared__ float smem[];
    float* As    = smem;                       // 64*256 = 16384 floats
    float* Bbuf0 = As + ROWS_PER_BLOCK * DIM;  // 16*256 floats
    float* Bbuf1 = Bbuf0 + BTILE * DIM;        // 16*256 floats

    const int t    = threadIdx.x;
    const int w    = t >> 5;       // wave in block (0..3)
    const int lane = t & 31;
    const int half = lane >> 4;    // 0: lanes 0-15, 1: lanes 16-31
    const int l16  = lane & 15;
    const int row0 = blockIdx.x * ROWS_PER_BLOCK;

    // Stage A tile (64 rows x 256 dims = 64KB) via async copy: 32 b128/thread
    {
        const float* src = X + (size_t)row0 * DIM;
#pragma unroll 8
        for (int j = 0; j < (ROWS_PER_BLOCK * DIM / 4) / 128; ++j)
            async_copy_b128(src + (t + j * 128) * 4, As + (t + j * 128) * 4);
    }
    // Prologue: async-stage B tile 0 into buffer 0 (8 b128/thread)
#pragma unroll
    for (int j = 0; j < (BTILE * DIM / 4) / 128; ++j)
        async_copy_b128(E + (t + j * 128) * 4, Bbuf0 + (t + j * 128) * 4);

    // Per-lane fragment base pointers (ISA f32 A/B VGPR striping):
    //   A 16x4: lane holds row = l16, dims {4*dk + 2*half, +1}
    //   B 4x16: lane holds col = l16, dims {4*dk + 2*half, +1}  (B = E^T tile)
    const float* pa  = As + (w * 16 + l16) * DIM + 2 * half;
    const float* pb0 = Bbuf0 + l16 * DIM + 2 * half;   // fixed base, buffer 0
    const float* pb1 = Bbuf1 + l16 * DIM + 2 * half;   // fixed base, buffer 1

    float minv[8];
    int   mini[8];
#pragma unroll
    for (int v = 0; v < 8; ++v) { minv[v] = 3.4e38f; mini[v] = 0; }

    // One pipeline stage: prefetch tile (kb+BTILE) into `nextdst` (if any),
    // drain older async transfers, sync, compute tile `kb` from `pb`.
    auto do_tile = [&](int kb, const float* pb, float* nextdst) {
        if (nextdst) {
            const float* src = E + (size_t)(kb + BTILE) * DIM;
#pragma unroll
            for (int j = 0; j < (BTILE * DIM / 4) / 128; ++j)
                async_copy_b128(src + (t + j * 128) * 4, nextdst + (t + j * 128) * 4);
            wait_asynccnt<(BTILE * DIM / 4) / 128>();   // only the 8 new in flight
        } else {
            wait_asynccnt<0>();
        }
        __syncthreads();   // all waves' portions of current tile visible

        const float e2   = enorm[kb + l16];   // per-lane, L2-resident
        const int   code = kb + l16;

        v8f c = {};
#pragma unroll 8
        for (int dk = 0; dk < DIM / 4; ++dk) {
            v2f a = *(const v2f*)(pa + 4 * dk);   // ds_load_2addr_b64 pairs
            v2f b = *(const v2f*)(pb + 4 * dk);
            c = __builtin_amdgcn_wmma_f32_16x16x4_f32(
                    false, a, false, b, (short)0, c, false, false);
        }

        // Running argmin on key = ||e||^2 - 2*dot  (||x||^2 constant per row)
#pragma unroll
        for (int v = 0; v < 8; ++v) {
            float key = e2 - 2.0f * c[v];
            if (key < minv[v]) { minv[v] = key; mini[v] = code; }  // '<' keeps lowest idx
        }
        __syncthreads();   // protect current buffer before it is re-targeted
    };

    for (int kb = 0; kb < NCODES; kb += 2 * BTILE) {
        do_tile(kb,         pb0, Bbuf1);                                     // even tile
        do_tile(kb + BTILE, pb1, (kb + 2 * BTILE < NCODES) ? Bbuf0 : nullptr); // odd tile
    }

    // Cross-lane argmin over the 16 columns held in each 16-lane group.
    // C-layout: VGPR v, lanes 0-15 -> row v; lanes 16-31 -> row v+8.
#pragma unroll
    for (int v = 0; v < 8; ++v) {
#pragma unroll
        for (int mask = 1; mask < 16; mask <<= 1) {
            float ov = __shfl_xor(minv[v], mask, 32);
            int   oi = __shfl_xor(mini[v], mask, 32);
            if (ov < minv[v] || (ov == minv[v] && oi < mini[v])) {
                minv[v] = ov; mini[v] = oi;
            }
        }
    }
    if (l16 == 0) {
#pragma unroll
        for (int v = 0; v < 8; ++v)
            idx_out[row0 + w * 16 + half * 8 + v] = mini[v];
    }
}

// ---------------------------------------------------------------------------
// Kernel 3: gather quantized rows + per-block SSE partial (deterministic)
// 1024 blocks x 256 threads x 32 floats = 8388608 elements
// ---------------------------------------------------------------------------
__global__ void __launch_bounds__(256)
vq_gather_kernel(const float* __restrict__ X, const float* __restrict__ E,
                 const int* __restrict__ idx, float* __restrict__ out,
                 float* __restrict__ partial) {
    const int t    = threadIdx.x;
    const int base = blockIdx.x * (256 * 32) + t * 32;  // 32 floats per thread
    const int row  = base >> 8;                          // /256, thread stays in one row
    const int dcol = base & 255;
    const int code = idx[row];

    const float4* xp = (const float4*)(X + base);
    const float4* ep = (const float4*)(E + (size_t)code * DIM + dcol);
    float4*       op = (float4*)(out + base);

    float s = 0.f;
#pragma unroll
    for (int j = 0; j < 8; ++j) {
        float4 q = ep[j];
        float4 x = xp[j];
        op[j] = q;
        float dx = x.x - q.x, dy = x.y - q.y, dz = x.z - q.z, dw = x.w - q.w;
        s += dx * dx + dy * dy + dz * dz + dw * dw;
    }

    // wave32 reduce, then block reduce (fixed order -> replay-deterministic)
#pragma unroll
    for (int mask = 1; mask < 32; mask <<= 1) s += __shfl_xor(s, mask, 32);
    __shared__ float red[8];
    if ((t & 31) == 0) red[t >> 5] = s;
    __syncthreads();
    if (t == 0) {
        float b = 0.f;
#pragma unroll
        for (int j = 0; j < 8; ++j) b += red[j];
        partial[blockIdx.x] = b;
    }
}

// ---------------------------------------------------------------------------
// Kernel 4: deterministic sum of 1024 partials -> losses
// ---------------------------------------------------------------------------
__global__ void __launch_bounds__(256)
vq_loss_kernel(const float* __restrict__ partial, float* __restrict__ losses) {
    __shared__ float red[256];
    float s = 0.f;
    for (int j = threadIdx.x; j < 1024; j += 256) s += partial[j];
    red[threadIdx.x] = s;
    __syncthreads();
    for (int step = 128; step > 0; step >>= 1) {
        if (threadIdx.x < step) red[threadIdx.x] += red[threadIdx.x + step];
        __syncthreads();
    }
    if (threadIdx.x == 0) {
        float mse = red[0] / (float)(NROWS * DIM);
        losses[0] = mse;          // codebook loss
        losses[1] = 0.25f * mse;  // commitment loss
    }
}

// ---------------------------------------------------------------------------
extern "C" void kernel_launch(void* const* d_in, const int* in_sizes, int n_in,
                              void* d_out, int out_size, void* d_ws, size_t ws_size,
                              hipStream_t stream) {
    const float* X = (const float*)d_in[0];   // (32,32,32,256) -> (32768, 256)
    const float* E = (const float*)d_in[1];   // (4096, 256)
    float*       out = (float*)d_out;         // 8388608 quantized + 2 losses

    float* enorm   = (float*)d_ws;            // 4096 f32
    int*   idx     = (int*)(enorm + NCODES);  // 32768 i32
    float* partial = (float*)(idx + NROWS);   // 1024 f32

    vq_enorm_kernel<<<NCODES / 256, 256, 0, stream>>>(E, enorm);

    size_t shmem = (size_t)(ROWS_PER_BLOCK * DIM + 2 * BTILE * DIM) * sizeof(float);
    vq_argmin_kernel<<<NROWS / ROWS_PER_BLOCK, 128, shmem, stream>>>(X, E, enorm, idx);

    vq_gather_kernel<<<1024, 256, 0, stream>>>(X, E, idx, out, partial);
    vq_loss_kernel<<<1, 256, 0, stream>>>(partial, out + (size_t)NROWS * DIM);
}